// ROIAlign_31885837205767
// MI455X (gfx1250) — compile-verified
//
#include <hip/hip_runtime.h>
#include <cstdint>

#define POOL     7
#define CHANNELS 256
#define GROUPS   4      // 64-lane groups per 256-thread block
#define LANES    64     // 64 lanes * float4 = 256 channels

__global__ __launch_bounds__(256) void roialign_async_kernel(
    const float* __restrict__ rois,
    const float* __restrict__ p2, const float* __restrict__ p3,
    const float* __restrict__ p4, const float* __restrict__ p5,
    const float* __restrict__ meta,
    float* __restrict__ out, int nroi)
{
    __shared__ float4 stage[GROUPS][4][LANES];   // 16 KB / block

    const int grp  = threadIdx.x >> 6;           // position group within block
    const int lane = threadIdx.x & (LANES - 1);  // channel slice (4 floats)
    const int P    = nroi * POOL * POOL;
    const int pos  = blockIdx.x * GROUPS + grp;
    if (pos >= P) return;

    const int n   = pos / (POOL * POOL);
    const int rem = pos - n * (POOL * POOL);
    const int py  = rem / POOL;
    const int px  = rem - py * POOL;

    // ROI columns: (batch, y1, x1, y2, x2) normalized
    const float bidf = rois[n * 5 + 0];
    const float y1   = rois[n * 5 + 1];
    const float x1   = rois[n * 5 + 2];
    const float y2   = rois[n * 5 + 3];
    const float x2   = rois[n * 5 + 4];

    // FPN level: clamp(4 + round(log2(sqrt(area)/224)), 2, 5); round = nearest-even (rintf)
    const float area = fmaxf(0.f, y2 - y1) * fmaxf(0.f, x2 - x1) * (meta[0] * meta[1]);
    const float lvlf = fminf(5.f, fmaxf(2.f, 4.f + rintf(log2f(sqrtf(area) * (1.0f / 224.0f)))));
    const int   lev  = (int)lvlf;

    const float* feat; int S;
    if      (lev == 2) { feat = p2; S = 256; }
    else if (lev == 3) { feat = p3; S = 128; }
    else if (lev == 4) { feat = p4; S = 64;  }
    else               { feat = p5; S = 32;  }

    const int   b   = (int)bidf;
    const float sm1 = (float)(S - 1);
    // crop_and_resize sample grid (n=7 > 1 branch)
    const float cy = y1 * sm1 + (float)py * ((y2 - y1) * sm1 * (1.0f / 6.0f));
    const float cx = x1 * sm1 + (float)px * ((x2 - x1) * sm1 * (1.0f / 6.0f));

    const float fy = floorf(cy), fx = floorf(cx);
    const float wy = cy - fy,    wx = cx - fx;
    const int iy0 = min(max((int)fy,     0), S - 1);
    const int iy1 = min(max((int)fy + 1, 0), S - 1);
    const int ix0 = min(max((int)fx,     0), S - 1);
    const int ix1 = min(max((int)fx + 1, 0), S - 1);
    const bool valid = (cy >= 0.f) && (cy <= sm1) && (cx >= 0.f) && (cx <= sm1);

    // Four gathered pixels; each lane owns a contiguous 16B channel slice.
    const size_t cslice = (size_t)lane * 4;
    const float* g00 = feat + ((size_t)((b * S + iy0) * S + ix0) * CHANNELS + cslice);
    const float* g01 = feat + ((size_t)((b * S + iy0) * S + ix1) * CHANNELS + cslice);
    const float* g10 = feat + ((size_t)((b * S + iy1) * S + ix0) * CHANNELS + cslice);
    const float* g11 = feat + ((size_t)((b * S + iy1) * S + ix1) * CHANNELS + cslice);

    // LDS byte addresses: on gfx1250 a generic LDS pointer carries the LDS
    // offset in addr[31:0], which is exactly what the async-LDS VDST wants.
    const uint32_t l0 = (uint32_t)(uintptr_t)&stage[grp][0][lane];
    const uint32_t l1 = l0 + LANES * (uint32_t)sizeof(float4);
    const uint32_t l2 = l1 + LANES * (uint32_t)sizeof(float4);
    const uint32_t l3 = l2 + LANES * (uint32_t)sizeof(float4);

    // CDNA5 async global->LDS copies (ASYNCcnt-tracked), then fence.
    asm volatile("global_load_async_to_lds_b128 %0, %1, off" :: "v"(l0), "v"(g00) : "memory");
    asm volatile("global_load_async_to_lds_b128 %0, %1, off" :: "v"(l1), "v"(g01) : "memory");
    asm volatile("global_load_async_to_lds_b128 %0, %1, off" :: "v"(l2), "v"(g10) : "memory");
    asm volatile("global_load_async_to_lds_b128 %0, %1, off" :: "v"(l3), "v"(g11) : "memory");
    asm volatile("s_wait_asynccnt 0" ::: "memory");

    const float4 v00 = stage[grp][0][lane];
    const float4 v01 = stage[grp][1][lane];
    const float4 v10 = stage[grp][2][lane];
    const float4 v11 = stage[grp][3][lane];

    float4 o;
    if (valid) {
        float t, bt;
        t = fmaf(wx, v01.x - v00.x, v00.x); bt = fmaf(wx, v11.x - v10.x, v10.x); o.x = fmaf(wy, bt - t, t);
        t = fmaf(wx, v01.y - v00.y, v00.y); bt = fmaf(wx, v11.y - v10.y, v10.y); o.y = fmaf(wy, bt - t, t);
        t = fmaf(wx, v01.z - v00.z, v00.z); bt = fmaf(wx, v11.z - v10.z, v10.z); o.z = fmaf(wy, bt - t, t);
        t = fmaf(wx, v01.w - v00.w, v00.w); bt = fmaf(wx, v11.w - v10.w, v10.w); o.w = fmaf(wy, bt - t, t);
    } else {
        o = make_float4(0.f, 0.f, 0.f, 0.f);
    }
    reinterpret_cast<float4*>(out + (size_t)pos * CHANNELS)[lane] = o;
}

extern "C" void kernel_launch(void* const* d_in, const int* in_sizes, int n_in,
                              void* d_out, int out_size, void* d_ws, size_t ws_size,
                              hipStream_t stream) {
    const float* rois = (const float*)d_in[0];
    const float* p2   = (const float*)d_in[1];
    const float* p3   = (const float*)d_in[2];
    const float* p4   = (const float*)d_in[3];
    const float* p5   = (const float*)d_in[4];
    const float* meta = (const float*)d_in[5];
    float* out = (float*)d_out;

    const int nroi   = in_sizes[0] / 5;
    const int P      = nroi * POOL * POOL;
    const int blocks = (P + GROUPS - 1) / GROUPS;
    roialign_async_kernel<<<blocks, 256, 0, stream>>>(rois, p2, p3, p4, p5, meta, out, nroi);
}